// GraphAttentionLayer_24249385353318
// MI455X (gfx1250) — compile-verified
//
#include <hip/hip_runtime.h>
#include <hip/hip_bf16.h>
#include <math.h>

// GAT layer, fused. N=8192, IN_F=512, OUT_F=64, leaky slope 0.2, elu alpha 1.
#define NN    8192
#define INF_  512
#define OUTF  64
#define SLOPE 0.2f

typedef float v2f __attribute__((ext_vector_type(2)));
typedef float v8f __attribute__((ext_vector_type(8)));

__device__ __forceinline__ float lrelu(float x) { return x >= 0.f ? x : SLOPE * x; }

// ---------------------------------------------------------------------------
// Kernel 1: h = input @ W   [8192x512]x[512x64], f32 WMMA 16x16x4.
// One wave per 16x16 output tile (2048 tiles), 8 waves/block -> 256 blocks.
// A layout: M=lane%16, K=vgpr+2*(lane/16). B layout mirrored. C rows r+8*(lane/16).
// ---------------------------------------------------------------------------
__global__ __launch_bounds__(256) void gat_gemm1(const float* __restrict__ in,
                                                 const float* __restrict__ W,
                                                 float* __restrict__ h) {
  const int wave = threadIdx.x >> 5;
  const int lane = threadIdx.x & 31;
  const int li = lane & 15;
  const int hi = lane >> 4;

  const int tile = blockIdx.x * 8 + wave;   // 0..2047
  const int i0 = (tile >> 2) * 16;          // row tile (512)
  const int n0 = (tile & 3) * 16;           // col tile (4)

  v8f acc = {};
  const float* arow = in + (size_t)(i0 + li) * INF_ + 2 * hi; // K base for this lane
  const float* bcol = W + (size_t)(2 * hi) * OUTF + n0 + li;

  for (int k0 = 0; k0 < INF_; k0 += 4) {
    v2f a, b;
    a.x = arow[k0];               // K = k0 + 2*hi
    a.y = arow[k0 + 1];           // K = k0 + 2*hi + 1
    b.x = bcol[(size_t)k0 * OUTF];          // W[k0+2*hi][n0+li]
    b.y = bcol[(size_t)k0 * OUTF + OUTF];   // W[k0+2*hi+1][n0+li]
    acc = __builtin_amdgcn_wmma_f32_16x16x4_f32(false, a, false, b,
                                                (short)0, acc, false, false);
  }
#pragma unroll
  for (int r = 0; r < 8; ++r)
    h[(size_t)(i0 + r + 8 * hi) * OUTF + n0 + li] = acc[r];
}

// ---------------------------------------------------------------------------
// Kernel 2: h1 = h @ a[:64], h2 = h @ a[64:]. One thread per row.
// ---------------------------------------------------------------------------
__global__ __launch_bounds__(256) void gat_attvec(const float* __restrict__ h,
                                                  const float* __restrict__ a,
                                                  float* __restrict__ h1,
                                                  float* __restrict__ h2) {
  const int r = blockIdx.x * blockDim.x + threadIdx.x;
  if (r >= NN) return;
  const float* hr = h + (size_t)r * OUTF;
  float s1 = 0.f, s2 = 0.f;
#pragma unroll 8
  for (int f = 0; f < OUTF; ++f) {
    float v = hr[f];
    s1 = fmaf(v, a[f], s1);
    s2 = fmaf(v, a[OUTF + f], s2);
  }
  h1[r] = s1;
  h2[r] = s2;
}

// ---------------------------------------------------------------------------
// Kernel 3: fused masked-softmax + aggregation + ELU.
// Block (8 waves) owns 16 rows; wave w streams j in [w*1024,(w+1)*1024):
//   p = adj>0 ? exp(lrelu(h1_i + h2_j)) : 1      (reference masks with 0!)
//   S[16x64] += p * h_j  via f32 WMMA 16x16x4 (A = P 16x4, B = h 4x16 tiles)
//   Z[i]     += p
// then LDS-reduce the 8 partials and write elu(S/Z).
// adj is streamed non-temporally (keep L2 for h) with prefetch ahead.
// ---------------------------------------------------------------------------
__global__ __launch_bounds__(256) void gat_fused(const float* __restrict__ h,
                                                 const int* __restrict__ adj,
                                                 const float* __restrict__ h1,
                                                 const float* __restrict__ h2,
                                                 float* __restrict__ out) {
  __shared__ float lh2[NN];            // 32 KB: all of h2
  __shared__ float lS[8][16 * OUTF];   // 32 KB: per-wave partial S
  __shared__ float lz[8][16];          // per-wave partial Z

  const int tid = threadIdx.x;
  const int wave = tid >> 5;
  const int lane = tid & 31;
  const int li = lane & 15;
  const int hi = lane >> 4;
  const int i0 = blockIdx.x * 16;      // 512 blocks

  for (int idx = tid; idx < NN; idx += 256) lh2[idx] = h2[idx];
  __syncthreads();

  const int jbase = wave * 1024;
  const float h1v = h1[i0 + li];                        // row of A for this lane
  const int* arow = adj + (size_t)(i0 + li) * NN;       // this lane's adj row
  float z = 0.f;
  v8f acc[4] = {};                                      // 4 feature tiles of 16

  for (int j0 = jbase; j0 < jbase + 1024; j0 += 4) {
    const int j = j0 + 2 * hi;                          // this lane's K pair
    if (((j0 - jbase) & 31) == 0)
      __builtin_prefetch(&arow[j + 128], 0, 0);         // global_prefetch_b8 ahead
    const int a0 = __builtin_nontemporal_load(&arow[j]);
    const int a1 = __builtin_nontemporal_load(&arow[j + 1]);
    const float p0 = (a0 > 0) ? __expf(lrelu(h1v + lh2[j]))     : 1.0f;
    const float p1 = (a1 > 0) ? __expf(lrelu(h1v + lh2[j + 1])) : 1.0f;
    z += p0 + p1;
    v2f A; A.x = p0; A.y = p1;

    const float* hb = h + (size_t)j * OUTF + li;        // h[j][li + t*16], next row +64
#pragma unroll
    for (int t = 0; t < 4; ++t) {
      v2f B;
      B.x = hb[t * 16];
      B.y = hb[OUTF + t * 16];
      acc[t] = __builtin_amdgcn_wmma_f32_16x16x4_f32(false, A, false, B,
                                                     (short)0, acc[t], false, false);
    }
  }

  // per-wave partials -> LDS. Row of C element (vgpr r) is r + 8*hi.
  float* Sp = &lS[wave][0];
#pragma unroll
  for (int t = 0; t < 4; ++t)
#pragma unroll
    for (int r = 0; r < 8; ++r)
      Sp[(r + 8 * hi) * OUTF + t * 16 + li] = acc[t][r];

  z += __shfl_xor(z, 16, 32);          // rows split across lane pairs (l, l+16)
  if (hi == 0) lz[wave][li] = z;
  __syncthreads();

  // reduce 8 partials, normalize, ELU, store. 1024 outputs / 256 threads.
  for (int e = tid; e < 16 * OUTF; e += 256) {
    const int row = e >> 6;
    float s = 0.f, Z = 0.f;
#pragma unroll
    for (int w = 0; w < 8; ++w) { s += lS[w][e]; Z += lz[w][row]; }
    float v = s / Z;
    v = (v > 0.f) ? v : expm1f(v);     // elu, alpha = 1
    out[(size_t)(i0 + row) * OUTF + (e & 63)] = v;
  }
}

// ---------------------------------------------------------------------------
extern "C" void kernel_launch(void* const* d_in, const int* in_sizes, int n_in,
                              void* d_out, int out_size, void* d_ws, size_t ws_size,
                              hipStream_t stream) {
  const float* inp = (const float*)d_in[0];   // [8192,512]
  const int*   adj = (const int*)  d_in[1];   // [8192,8192]
  const float* W   = (const float*)d_in[2];   // [512,64]
  const float* a   = (const float*)d_in[3];   // [128,1]
  float* out = (float*)d_out;                 // [8192,64]

  float* h  = (float*)d_ws;                   // 8192*64 f32 = 2 MB
  float* h1 = h + (size_t)NN * OUTF;          // 8192 f32
  float* h2 = h1 + NN;                        // 8192 f32

  gat_gemm1 <<<256, 256, 0, stream>>>(inp, W, h);
  gat_attvec<<<32,  256, 0, stream>>>(h, a, h1, h2);
  gat_fused <<<512, 256, 0, stream>>>(h, adj, h1, h2, out);
}